// CoRelation_58411555226212
// MI455X (gfx1250) — compile-verified
//
#include <hip/hip_runtime.h>
#include <math.h>

// Problem constants
#define CCH   1024
#define RC    64
#define L7    49
#define HW8   64
#define BZ    4
#define BP    1024
#define PPB   8                  // proposals per block (WMMA M-tile padded to 16)
#define NBLK  (BP / PPB)         // 128 blocks -> fills the WGP array
#define THREADS 256

typedef float v2f __attribute__((ext_vector_type(2)));
typedef float v8f __attribute__((ext_vector_type(8)));

// workspace layout (float offsets)
#define OFF_RQ    0              // [4][1024][49]
#define OFF_QFEAT 200704         // [4][64][49]
#define OFF_QBAR  213248         // [4][49]
#define OFF_SUMQ  213444         // [4][64]
#define OFF_SQ    213700         // [4]
#define OFF_WBAR  213704         // [1024]
#define OFF_BBAR  214728         // [1]

// output layout (float offsets)
#define OUT_FQUERY 51380224
#define OUT_PATT   (51380224 + 67108864)

// ---------------- prologue 1: bilinear resize 8x8 -> 7x7, half-pixel centers
__global__ void prep_resize(const float* __restrict__ xq, float* __restrict__ ws) {
  int bc = blockIdx.x;                 // b*1024 + c, 0..4095
  int l  = threadIdx.x;
  if (l >= L7) return;
  int oy = l / 7, ox = l % 7;
  const float scale = 8.0f / 7.0f;
  float sy = (oy + 0.5f) * scale - 0.5f;
  float sx = (ox + 0.5f) * scale - 0.5f;
  float y0f = floorf(sy), x0f = floorf(sx);
  float fy = sy - y0f, fx = sx - x0f;
  int y0 = (int)y0f, x0 = (int)x0f;
  int y1 = y0 + 1 > 7 ? 7 : y0 + 1;
  int x1 = x0 + 1 > 7 ? 7 : x0 + 1;
  y0 = y0 < 0 ? 0 : y0; x0 = x0 < 0 ? 0 : x0;
  const float* src = xq + (size_t)bc * HW8;
  float v00 = src[y0*8+x0], v01 = src[y0*8+x1];
  float v10 = src[y1*8+x0], v11 = src[y1*8+x1];
  float v = (1.f-fy)*((1.f-fx)*v00 + fx*v01) + fy*((1.f-fx)*v10 + fx*v11);
  ws[OFF_RQ + (size_t)bc * L7 + l] = v;
}

// ---------------- prologue 2: q_feat[b][s][l] = qc_w[s,:] . r_query[b,:,l] + qc_b[s]
__global__ void prep_qfeat(const float* __restrict__ qc_w, const float* __restrict__ qc_b,
                           float* __restrict__ ws) {
  int bs = blockIdx.x;                 // b*64 + s, 0..255
  int l  = threadIdx.x;
  if (l >= L7) return;
  int b = bs >> 6, s = bs & 63;
  const float* rq = ws + OFF_RQ + (size_t)b * CCH * L7 + l;
  const float* w  = qc_w + (size_t)s * CCH;
  float acc = qc_b[s];
  for (int c = 0; c < CCH; ++c) acc += w[c] * rq[(size_t)c * L7];
  ws[OFF_QFEAT + (size_t)bs * L7 + l] = acc;
}

// ---------------- prologue 3: wbar, bbar, qbar, sumq, Sq  (one block, 1024 threads)
__global__ void prep_stats(const float* __restrict__ pc_w, const float* __restrict__ pc_b,
                           float* __restrict__ ws) {
  int tid = threadIdx.x;
  { // wbar[c] = mean_r pc_w[r][c]
    float s = 0.f;
    for (int r = 0; r < RC; ++r) s += pc_w[r * CCH + tid];
    ws[OFF_WBAR + tid] = s * (1.0f / RC);
  }
  if (tid == 0) {
    float s = 0.f;
    for (int r = 0; r < RC; ++r) s += pc_b[r];
    ws[OFF_BBAR] = s * (1.0f / RC);
  }
  const float* qf = ws + OFF_QFEAT;
  if (tid < BZ * L7) {  // qbar[b][l] = mean_s qfeat
    int b = tid / L7, l = tid % L7;
    float s = 0.f;
    for (int ss = 0; ss < RC; ++ss) s += qf[(b * RC + ss) * L7 + l];
    ws[OFF_QBAR + tid] = s * (1.0f / RC);
  }
  if (tid < BZ * RC) {  // sumq[b][s] = sum_l qfeat
    int b = tid >> 6, ss = tid & 63;
    float s = 0.f;
    for (int l = 0; l < L7; ++l) s += qf[(b * RC + ss) * L7 + l];
    ws[OFF_SUMQ + tid] = s;
  }
  __syncthreads();
  if (tid < BZ) {       // Sq[b] = sum_l qbar
    float s = 0.f;
    for (int l = 0; l < L7; ++l) s += ws[OFF_QBAR + tid * L7 + l];
    ws[OFF_SQ + tid] = s;
  }
}

// ---------------- main fused kernel: 128 blocks, 8 proposals each
__global__ __launch_bounds__(THREADS)
void corel_main(const float* __restrict__ x_props, const float* __restrict__ x_query,
                const float* __restrict__ pc_w,   const float* __restrict__ pc_b,
                const float* __restrict__ pm_w,   const float* __restrict__ pm_b,
                const float* __restrict__ qm_w,   const float* __restrict__ qm_b,
                const float* __restrict__ ws,     float* __restrict__ out) {
  // WMMA operand buffers keep 16 rows (M-tile); rows >= PPB are zero-padded.
  __shared__ __attribute__((aligned(16))) float t_lds[16 * CCH];      // t, reused as p_att
  __shared__ __attribute__((aligned(16))) float qatt_lds[PPB * CCH];  // q_att (valid rows only)
  __shared__ __attribute__((aligned(16))) float qfeat_s[RC * L7];
  __shared__ __attribute__((aligned(16))) float wbar_s[CCH];
  __shared__ __attribute__((aligned(16))) float qbar_s[52];
  __shared__ __attribute__((aligned(16))) float sumq_s[RC];
  __shared__ __attribute__((aligned(16))) float xbar_lds[PPB * 52];
  __shared__ __attribute__((aligned(16))) float prel_lds[16 * RC];
  __shared__ __attribute__((aligned(16))) float qrel_lds[16 * RC];
  __shared__ float scal[2];   // Sq[b], bbar

  const int tid   = threadIdx.x;
  const int blk   = blockIdx.x;
  const int b     = blk >> 5;          // image index (32 blocks per image)
  const int pbase = blk * PPB;         // first proposal

  for (int i = tid; i < RC * L7; i += THREADS) qfeat_s[i] = ws[OFF_QFEAT + b * (RC * L7) + i];
  for (int i = tid; i < CCH;     i += THREADS) wbar_s[i]  = ws[OFF_WBAR + i];
  if (tid < L7) qbar_s[tid] = ws[OFF_QBAR + b * L7 + tid];
  else if (tid < 52) qbar_s[tid] = 0.f;
  if (tid < RC) sumq_s[tid] = ws[OFF_SUMQ + b * RC + tid];
  for (int i = tid; i < PPB * 52; i += THREADS) xbar_lds[i] = 0.f;
  // zero pad rows PPB..15 of WMMA A-operand buffers
  for (int i = PPB * CCH + tid; i < 16 * CCH; i += THREADS) t_lds[i] = 0.f;
  for (int i = PPB * RC + tid; i < 16 * RC; i += THREADS) qrel_lds[i] = 0.f;
  if (tid == 0) scal[0] = ws[OFF_SQ + b];
  if (tid == 1) scal[1] = ws[OFF_BBAR];
  __syncthreads();

  const int lane = tid & 31;

  // ---- Phase A: stream x_props once; t[pp][c] and xbar[pp][l]
  for (int pp = 0; pp < PPB; ++pp) {
    const float* xp = x_props + (size_t)(pbase + pp) * (CCH * L7);
    float xacc[L7];
    #pragma unroll
    for (int l = 0; l < L7; ++l) xacc[l] = 0.f;
    #pragma unroll
    for (int k = 0; k < 4; ++k) {
      int c = tid + k * THREADS;
      const float* row = xp + c * L7;
      if (k < 3) __builtin_prefetch(xp + (c + THREADS) * L7, 0, 3);
      float wb = wbar_s[c];
      float tv = 0.f;
      #pragma unroll
      for (int l = 0; l < L7; ++l) {
        float v = row[l];
        tv += v * qbar_s[l];
        xacc[l] += wb * v;
      }
      t_lds[pp * CCH + c] = tv;
    }
    #pragma unroll
    for (int l = 0; l < L7; ++l) {
      float v = xacc[l];
      v += __shfl_xor(v, 16, 32);
      v += __shfl_xor(v,  8, 32);
      v += __shfl_xor(v,  4, 32);
      v += __shfl_xor(v,  2, 32);
      v += __shfl_xor(v,  1, 32);
      if (lane == 0) atomicAdd(&xbar_lds[pp * 52 + l], v);
    }
  }
  __syncthreads();

  const float Sq   = scal[0];
  const float bbar = scal[1];
  const int wave = tid >> 5;
  const int hi   = lane >> 4;     // half-wave select (K pair / M+8 row)
  const int lm   = lane & 15;     // A row m / B,D column n

  // ---- Phase B: waves 0-3: GEMM1 p_rel = t @ pc_w^T via WMMA f32 16x16x4
  if (wave < 4) {
    const int nb = wave * 16;                 // r-tile base
    v8f acc = {};
    for (int kb = 0; kb < CCH; kb += 4) {
      v2f a  = *(const v2f*)&t_lds[lm * CCH + kb + 2 * hi];
      v2f bf = *(const v2f*)&pc_w[(size_t)(nb + lm) * CCH + kb + 2 * hi];
      acc = __builtin_amdgcn_wmma_f32_16x16x4_f32(false, a, false, bf,
                                                  (short)0, acc, false, false);
    }
    int r = nb + lm;
    float bias = pc_b[r] * Sq;
    #pragma unroll
    for (int v = 0; v < 8; ++v)
      prel_lds[(v + 8 * hi) * RC + r] = acc[v] + bias;
  } else {
    // waves 4-7: q_rel[pp][s] = xbar[pp,:] . qfeat[s,:] + bbar*sumq[s]
    int t2 = tid - 128;
    for (int k = 0; k < 4; ++k) {
      int j = t2 + k * 128;                   // 0..511
      int pp = j >> 6, s = j & 63;
      float acc = bbar * sumq_s[s];
      #pragma unroll
      for (int l = 0; l < L7; ++l)
        acc += xbar_lds[pp * 52 + l] * qfeat_s[s * L7 + l];
      qrel_lds[pp * RC + s] = acc;
    }
  }
  __syncthreads();

  // ---- GEMM2 (both gates): att[PPB][1024] = sigmoid(rel[PPB][64] @ W^T + bias)
  #pragma unroll
  for (int g = 0; g < 2; ++g) {
    const float* W  = (g == 0) ? pm_w : qm_w;
    const float* Bb = (g == 0) ? pm_b : qm_b;
    v2f afr[16];
    #pragma unroll
    for (int kk = 0; kk < 16; ++kk) {
      afr[kk] = (g == 0) ? *(const v2f*)&prel_lds[lm * RC + kk * 4 + 2 * hi]
                         : *(const v2f*)&qrel_lds[lm * RC + kk * 4 + 2 * hi];
    }
    for (int nt = 0; nt < 8; ++nt) {
      int cc = (wave * 8 + nt) * 16 + lm;     // output channel
      v8f acc = {};
      #pragma unroll
      for (int kk = 0; kk < 16; ++kk) {
        v2f bf = *(const v2f*)&W[(size_t)cc * RC + kk * 4 + 2 * hi];
        acc = __builtin_amdgcn_wmma_f32_16x16x4_f32(false, afr[kk], false, bf,
                                                    (short)0, acc, false, false);
      }
      float bias = Bb[cc];
      if (hi == 0) {                          // rows 0..7 are the valid proposals
        #pragma unroll
        for (int v = 0; v < 8; ++v) {
          float z = acc[v] + bias;
          float sg = 1.0f / (1.0f + __expf(-z));
          if (g == 0) t_lds[v * CCH + cc] = sg;      // p_att (reuse t rows 0..7)
          else        qatt_lds[v * CCH + cc] = sg;   // q_att
        }
      }
    }
  }
  __syncthreads();

  // ---- Phase C: fully coalesced application + outputs
  { // p_att output [bp][c]
    float* o3 = out + OUT_PATT + (size_t)pbase * CCH;
    for (int i = tid; i < PPB * CCH; i += THREADS) o3[i] = t_lds[i];
  }
  { // f_props = x_props * p_att (x_props reread is L2-hot)
    const float* src = x_props + (size_t)pbase * (CCH * L7);
    float* o1 = out + (size_t)pbase * (CCH * L7);
    const int N = PPB * CCH * L7;
    for (int i = tid; i < N; i += THREADS) {
      unsigned ui = (unsigned)i;
      unsigned pp = ui / (CCH * L7);
      unsigned c  = (ui - pp * (CCH * L7)) / L7;
      o1[i] = src[i] * t_lds[pp * CCH + c];
    }
  }
  { // f_query = broadcast(x_query) * q_att
    const float* xq = x_query + (size_t)b * (CCH * HW8);
    float* o2 = out + OUT_FQUERY + (size_t)pbase * (CCH * HW8);
    const int N = PPB * CCH * HW8;
    for (int i = tid; i < N; i += THREADS) {
      int pp  = i >> 16;          // / (1024*64)
      int chw = i & 65535;
      o2[i] = xq[chw] * qatt_lds[pp * CCH + (chw >> 6)];
    }
  }
}

extern "C" void kernel_launch(void* const* d_in, const int* in_sizes, int n_in,
                              void* d_out, int out_size, void* d_ws, size_t ws_size,
                              hipStream_t stream) {
  (void)in_sizes; (void)n_in; (void)out_size; (void)ws_size;
  const float* x_props = (const float*)d_in[0];
  const float* x_query = (const float*)d_in[1];
  const float* pc_w    = (const float*)d_in[2];
  const float* pc_b    = (const float*)d_in[3];
  const float* qc_w    = (const float*)d_in[4];
  const float* qc_b    = (const float*)d_in[5];
  const float* pm_w    = (const float*)d_in[6];
  const float* pm_b    = (const float*)d_in[7];
  const float* qm_w    = (const float*)d_in[8];
  const float* qm_b    = (const float*)d_in[9];
  float* out = (float*)d_out;
  float* ws  = (float*)d_ws;

  prep_resize<<<BZ * CCH, 64, 0, stream>>>(x_query, ws);
  prep_qfeat<<<BZ * RC, 64, 0, stream>>>(qc_w, qc_b, ws);
  prep_stats<<<1, 1024, 0, stream>>>(pc_w, pc_b, ws);
  corel_main<<<NBLK, THREADS, 0, stream>>>(x_props, x_query, pc_w, pc_b,
                                           pm_w, pm_b, qm_w, qm_b, ws, out);
}